// GNN_13134009991659
// MI455X (gfx1250) — compile-verified
//
#include <hip/hip_runtime.h>

#define N_NODES 50000
#define N_EDGES 600000
#define HID 128
#define N_LAYERS 5
#define ROW_TILES ((N_NODES + 15) / 16)   // 3125

typedef __attribute__((ext_vector_type(2))) float v2f;
typedef __attribute__((ext_vector_type(4))) float v4f;
typedef __attribute__((ext_vector_type(8))) float v8f;

// ---------------------------------------------------------------- utilities

__global__ __launch_bounds__(256) void zero_f32(float* __restrict__ p, int n) {
    int i = blockIdx.x * 256 + threadIdx.x;
    if (i < n) p[i] = 0.0f;
}

__global__ __launch_bounds__(256) void deg_count(const long long* __restrict__ dst,
                                                 float* __restrict__ deg) {
    int e = blockIdx.x * 256 + threadIdx.x;
    if (e < N_EDGES) atomicAdd(&deg[(int)dst[e]], 1.0f);
}

__global__ __launch_bounds__(256) void deg_invert(const float* __restrict__ deg,
                                                  float* __restrict__ deg_inv) {
    int i = blockIdx.x * 256 + threadIdx.x;
    if (i < N_NODES) deg_inv[i] = 1.0f / fmaxf(deg[i], 1.0f);
}

// ------------------------------------------------- edge-parallel scatter-add
// One thread handles one float4 (4 channels) of one edge: coalesced 16B loads
// of h[src], 4 f32 L2 atomics into agg[dst]. Whole h/agg are L2-resident.

__global__ __launch_bounds__(256) void scatter_add(const float* __restrict__ h,
                                                   const long long* __restrict__ src,
                                                   const long long* __restrict__ dst,
                                                   float* __restrict__ agg) {
    unsigned idx = blockIdx.x * 256u + threadIdx.x;
    unsigned e = idx >> 5;        // edge id
    unsigned q = idx & 31u;       // which float4 of the 128-wide row
    if (e >= N_EDGES) return;
    int s = (int)src[e];
    int d = (int)dst[e];
    v4f v = ((const v4f*)(h + (size_t)s * HID))[q];
    float* p = agg + (size_t)d * HID + (size_t)q * 4;
    atomicAdd(p + 0, v[0]);
    atomicAdd(p + 1, v[1]);
    atomicAdd(p + 2, v[2]);
    atomicAdd(p + 3, v[3]);
}

// ------------------------------------------------------- fused SAGE GEMM
// out[m,n] = relu( sum_k (agg[m,k]*dinv[m]) * Wl[n,k]
//                + sum_k  h[m,k]            * Wr[n,k] + b[n] )
// One wave per 16-row tile, covering all 128 output columns with 8
// V_WMMA_F32_16X16X4_F32 accumulators (full fp32 precision — compute is
// negligible vs the L2-bound scatter phase, so no need to downcast).
//
// f32 16x16x4 fragment layouts (wave32):
//   A (16x4):  lane 0-15 -> M=lane,    VGPR0=K0, VGPR1=K1
//              lane16-31 -> M=lane-16, VGPR0=K2, VGPR1=K3
//   B (4x16):  lane 0-15 -> N=lane,    VGPR0=K0, VGPR1=K1
//              lane16-31 -> N=lane-16, VGPR0=K2, VGPR1=K3
//   C/D (16x16): VGPR v -> M = v + (lane<16 ? 0 : 8), N = lane&15
// Since B[k][n] = W[n][k], each lane's B fragment is 2 contiguous floats of
// row n of W — no transpose needed.

__global__ __launch_bounds__(128) void sage_gemm(const float* __restrict__ agg,
                                                 const float* __restrict__ h,
                                                 const float* __restrict__ Wl,
                                                 const float* __restrict__ Wr,
                                                 const float* __restrict__ bias,
                                                 const float* __restrict__ deg_inv,
                                                 float* __restrict__ out,
                                                 int relu) {
    const int wave = threadIdx.x >> 5;
    const int lane = threadIdx.x & 31;
    const int tile = blockIdx.x * 4 + wave;
    if (tile >= ROW_TILES) return;              // wave-uniform: EXEC stays all-1s
    const int row0 = tile << 4;
    const int r = lane & 15;                    // A-row / B-col within tile
    const int kb = (lane < 16) ? 0 : 2;         // K sub-offset for this half-wave
    const int row = row0 + r;
    const float dinv = deg_inv[row];

    v8f acc[8];
#pragma unroll
    for (int nt = 0; nt < 8; ++nt) acc[nt] = (v8f){0.f,0.f,0.f,0.f,0.f,0.f,0.f,0.f};

    const float* aggrow = agg + (size_t)row * HID;
    const float* hrow   = h   + (size_t)row * HID;

    // phase 1: mean @ Wl^T   (scale by deg_inv on the fly)
#pragma unroll 4
    for (int ks = 0; ks < HID; ks += 4) {
        v2f a;
        a[0] = aggrow[ks + kb + 0] * dinv;
        a[1] = aggrow[ks + kb + 1] * dinv;
#pragma unroll
        for (int nt = 0; nt < 8; ++nt) {
            const float* wb = Wl + (size_t)(nt * 16 + r) * HID + ks + kb;
            v2f bf;
            bf[0] = wb[0];
            bf[1] = wb[1];
            acc[nt] = __builtin_amdgcn_wmma_f32_16x16x4_f32(
                false, a, false, bf, (short)0, acc[nt], false, false);
        }
    }

    // phase 2: h @ Wr^T
#pragma unroll 4
    for (int ks = 0; ks < HID; ks += 4) {
        v2f a;
        a[0] = hrow[ks + kb + 0];
        a[1] = hrow[ks + kb + 1];
#pragma unroll
        for (int nt = 0; nt < 8; ++nt) {
            const float* wb = Wr + (size_t)(nt * 16 + r) * HID + ks + kb;
            v2f bf;
            bf[0] = wb[0];
            bf[1] = wb[1];
            acc[nt] = __builtin_amdgcn_wmma_f32_16x16x4_f32(
                false, a, false, bf, (short)0, acc[nt], false, false);
        }
    }

    // epilogue: + bias, optional relu, store fp32
    const int mbase = row0 + ((lane < 16) ? 0 : 8);
#pragma unroll
    for (int nt = 0; nt < 8; ++nt) {
        const int n = nt * 16 + r;
        const float bv = bias[n];
#pragma unroll
        for (int v = 0; v < 8; ++v) {
            float o = acc[nt][v] + bv;
            if (relu) o = fmaxf(o, 0.0f);
            out[(size_t)(mbase + v) * HID + n] = o;
        }
    }
}

// ---------------------------------------------------------------- driver

extern "C" void kernel_launch(void* const* d_in, const int* in_sizes, int n_in,
                              void* d_out, int out_size, void* d_ws, size_t ws_size,
                              hipStream_t stream) {
    const float*     x  = (const float*)d_in[0];                 // [N, 128]
    const long long* ei = (const long long*)d_in[1];             // [2, E] int64
    const float*     Wl = (const float*)d_in[2];                 // [5,128,128]
    const float*     Wr = (const float*)d_in[3];                 // [5,128,128]
    const float*     b  = (const float*)d_in[4];                 // [5,128]
    const long long* src = ei;
    const long long* dst = ei + N_EDGES;

    const size_t FEAT = (size_t)N_NODES * HID;                   // 6.4M floats
    float* bufA    = (float*)d_ws;                               // 25.6 MB
    float* bufB    = bufA + FEAT;                                // 25.6 MB
    float* deg     = bufB + FEAT;                                // 200 KB
    float* deg_inv = deg + N_NODES;                              // 200 KB

    // degree (once)
    zero_f32<<<(N_NODES + 255) / 256, 256, 0, stream>>>(deg, N_NODES);
    deg_count<<<(N_EDGES + 255) / 256, 256, 0, stream>>>(dst, deg);
    deg_invert<<<(N_NODES + 255) / 256, 256, 0, stream>>>(deg, deg_inv);

    const int zero_blocks    = ((int)FEAT + 255) / 256;          // 25000
    const int scatter_blocks = (N_EDGES * 32 + 255) / 256;       // 75000
    const int gemm_blocks    = (ROW_TILES + 3) / 4;              // 782

    for (int l = 0; l < N_LAYERS; ++l) {
        const float* hin = (l == 0) ? x : ((l & 1) ? bufA : bufB);
        float* agg       = (l & 1) ? bufB : bufA;
        float* outp      = (l == N_LAYERS - 1) ? (float*)d_out : agg; // in-place over agg
        zero_f32<<<zero_blocks, 256, 0, stream>>>(agg, (int)FEAT);
        scatter_add<<<scatter_blocks, 256, 0, stream>>>(hin, src, dst, agg);
        sage_gemm<<<gemm_blocks, 128, 0, stream>>>(
            agg, hin, Wl + (size_t)l * HID * HID, Wr + (size_t)l * HID * HID,
            b + (size_t)l * HID, deg_inv, outp, (l < N_LAYERS - 1) ? 1 : 0);
    }
}